// GNN_61418032333092
// MI455X (gfx1250) — compile-verified
//
#include <hip/hip_runtime.h>
#include <hip/hip_bf16.h>
#include <math.h>

typedef __attribute__((ext_vector_type(2))) float v2f;
typedef __attribute__((ext_vector_type(8))) float v8f;

// ---------------------------------------------------------------------------
// Small utility kernels
// ---------------------------------------------------------------------------

__global__ void gin_zero_f4(float4* __restrict__ p, long long n4) {
  long long i = (long long)blockIdx.x * blockDim.x + threadIdx.x;
  if (i < n4) p[i] = make_float4(0.f, 0.f, 0.f, 0.f);
}

__global__ void gin_zero_i(int* __restrict__ p, int n) {
  int i = blockIdx.x * blockDim.x + threadIdx.x;
  if (i < n) p[i] = 0;
}

// ---------------------------------------------------------------------------
// CSR-by-destination build (replaces 358M float atomics with 1.6M int atomics)
// ---------------------------------------------------------------------------

__global__ void gin_degree(const int* __restrict__ dst, int* __restrict__ deg,
                           int n_edges) {
  int e = blockIdx.x * blockDim.x + threadIdx.x;
  if (e < n_edges) atomicAdd(&deg[dst[e]], 1);
}

// per-block exclusive scan of deg -> rowptr (block-local), block totals out
__global__ __launch_bounds__(256) void gin_scan_block(
    const int* __restrict__ deg, int* __restrict__ out,
    int* __restrict__ partials, int n) {
  __shared__ int sm[256];
  const int t = threadIdx.x;
  const int i = blockIdx.x * 256 + t;
  const int v = (i < n) ? deg[i] : 0;
  sm[t] = v;
  __syncthreads();
#pragma unroll
  for (int off = 1; off < 256; off <<= 1) {
    const int add = (t >= off) ? sm[t - off] : 0;
    __syncthreads();
    sm[t] += add;
    __syncthreads();
  }
  if (i < n) out[i] = sm[t] - v;  // exclusive
  if (t == 255) partials[blockIdx.x] = sm[255];
}

// single-block exclusive scan of the block totals (nb <= 256)
__global__ __launch_bounds__(256) void gin_scan_partials(int* __restrict__ p,
                                                         int nb) {
  __shared__ int sm[256];
  const int t = threadIdx.x;
  const int v = (t < nb) ? p[t] : 0;
  sm[t] = v;
  __syncthreads();
#pragma unroll
  for (int off = 1; off < 256; off <<= 1) {
    const int add = (t >= off) ? sm[t - off] : 0;
    __syncthreads();
    sm[t] += add;
    __syncthreads();
  }
  if (t < nb) p[t] = sm[t] - v;
}

__global__ void gin_scan_finalize(int* __restrict__ rowptr,
                                  const int* __restrict__ partials,
                                  int* __restrict__ cursor, int n,
                                  int n_edges) {
  int i = blockIdx.x * blockDim.x + threadIdx.x;
  if (i < n) {
    const int r = rowptr[i] + partials[i >> 8];
    rowptr[i] = r;
    cursor[i] = r;
  }
  if (i == 0) rowptr[n] = n_edges;
}

__global__ void gin_csr_fill(const int* __restrict__ src,
                             const int* __restrict__ dst,
                             int* __restrict__ cursor, int* __restrict__ csr,
                             int n_edges) {
  int e = blockIdx.x * blockDim.x + threadIdx.x;
  if (e < n_edges) {
    const int pos = atomicAdd(&cursor[dst[e]], 1);
    csr[pos] = src[e];
  }
}

// ---------------------------------------------------------------------------
// Fused aggregation + (1+eps)*x: gather-sum over incoming edges (atomic-free).
// One wave per node; all lanes broadcast-read csr[j], gather from L2.
//   D==128: lane owns 4 consecutive features (float4 path)
//   D==32 : lane owns 1 feature
// ---------------------------------------------------------------------------
template <int D>
__global__ __launch_bounds__(128) void gin_aggregate(
    const float* __restrict__ x, const int* __restrict__ rowptr,
    const int* __restrict__ csr, const float* __restrict__ epsp,
    float* __restrict__ h, int n_nodes) {
  const int node = blockIdx.x * 4 + (threadIdx.x >> 5);
  const int lane = threadIdx.x & 31;
  if (node >= n_nodes) return;  // uniform per wave
  const int beg = rowptr[node];
  const int end = rowptr[node + 1];
  const float se = 1.0f + epsp[0];
  if constexpr (D == 128) {
    const int c0 = lane * 4;
    float4 acc = make_float4(0.f, 0.f, 0.f, 0.f);
    for (int j = beg; j < end; ++j) {
      const int s = csr[j];  // same addr in all lanes -> broadcast
      const float4 v = *(const float4*)(x + (size_t)s * 128 + c0);
      acc.x += v.x;
      acc.y += v.y;
      acc.z += v.z;
      acc.w += v.w;
    }
    const float4 xv = *(const float4*)(x + (size_t)node * 128 + c0);
    float4 r;
    r.x = fmaf(se, xv.x, acc.x);
    r.y = fmaf(se, xv.y, acc.y);
    r.z = fmaf(se, xv.z, acc.z);
    r.w = fmaf(se, xv.w, acc.w);
    *(float4*)(h + (size_t)node * 128 + c0) = r;
  } else {  // D == 32
    float acc = 0.f;
    for (int j = beg; j < end; ++j) acc += x[(size_t)csr[j] * 32 + lane];
    h[(size_t)node * 32 + lane] =
        fmaf(se, x[(size_t)node * 32 + lane], acc);
  }
}

// ---------------------------------------------------------------------------
// GIN MLP: Linear(D_IN->D_HID) -> BN(eval) -> ReLU -> Linear(D_HID->D_OUT)
// One wave32 per 16-node tile, V_WMMA_F32_16X16X4_F32 for both GEMMs (exact
// fp32). C/D-layout -> A-layout transpose between GEMMs goes through LDS.
// ---------------------------------------------------------------------------
template <int D_IN, int D_HID, int D_OUT>
__global__ __launch_bounds__(32) void gin_mlp(const float* __restrict__ h,
                                              const float* __restrict__ W1,
                                              const float* __restrict__ b1,
                                              const float* __restrict__ gamma,
                                              const float* __restrict__ beta,
                                              const float* __restrict__ mean,
                                              const float* __restrict__ var,
                                              const float* __restrict__ W2,
                                              const float* __restrict__ b2,
                                              float* __restrict__ out,
                                              int n_nodes) {
  constexpr int NT1 = D_HID / 16;
  constexpr int NT2 = D_OUT / 16;
  __shared__ float lds[16 * D_HID];

  const int lane = threadIdx.x;  // 0..31, wave32
  const int m = lane & 15;       // A row / B col / C col
  const int kh = lane >> 4;      // K-pair select (A/B), row-half (C/D)
  const int row0 = blockIdx.x * 16;

  // ---- GEMM1: acc1 = h_tile(16 x D_IN) @ W1(D_IN x D_HID) -----------------
  v8f acc1[NT1];
#pragma unroll
  for (int nt = 0; nt < NT1; ++nt) acc1[nt] = (v8f){};

#pragma unroll 4
  for (int k0 = 0; k0 < D_IN; k0 += 4) {
    const int ka = k0 + 2 * kh;
    const int rr = min(row0 + m, n_nodes - 1);  // keep EXEC all-ones
    const float* hp = h + (size_t)rr * D_IN + ka;
    v2f a;
    a.x = hp[0];
    a.y = hp[1];
#pragma unroll
    for (int nt = 0; nt < NT1; ++nt) {
      const int col = nt * 16 + m;
      v2f b;
      b.x = W1[(size_t)ka * D_HID + col];
      b.y = W1[(size_t)(ka + 1) * D_HID + col];
      acc1[nt] = __builtin_amdgcn_wmma_f32_16x16x4_f32(
          false, a, false, b, (short)0, acc1[nt], false, false);
    }
  }

  // ---- Epilogue 1: bias + BN(eval) + ReLU, spill to LDS row-major ---------
#pragma unroll
  for (int nt = 0; nt < NT1; ++nt) {
    const int col = nt * 16 + m;
    const float bb = b1[col];
    const float mu = mean[col];
    const float sc = gamma[col] * rsqrtf(var[col] + 1e-5f);
    const float bt = beta[col];
#pragma unroll
    for (int v = 0; v < 8; ++v) {
      const int r = v + 8 * kh;  // C/D layout: VGPR v + lane-half -> row
      float val = acc1[nt][v] + bb;
      val = fmaf(val - mu, sc, bt);
      val = fmaxf(val, 0.0f);
      lds[r * D_HID + col] = val;
    }
  }
  // Cross-lane LDS RAW (transpose): drain DS pipe before re-reading.
  __asm__ volatile("s_wait_dscnt 0" ::: "memory");

  // ---- GEMM2: acc2 = hidden(16 x D_HID) @ W2(D_HID x D_OUT) ---------------
  v8f acc2[NT2];
#pragma unroll
  for (int nt = 0; nt < NT2; ++nt) acc2[nt] = (v8f){};

#pragma unroll
  for (int k0 = 0; k0 < D_HID; k0 += 4) {
    const int ka = k0 + 2 * kh;
    v2f a;
    a.x = lds[m * D_HID + ka];
    a.y = lds[m * D_HID + ka + 1];
#pragma unroll
    for (int nt = 0; nt < NT2; ++nt) {
      const int col = nt * 16 + m;
      v2f b;
      b.x = W2[(size_t)ka * D_OUT + col];
      b.y = W2[(size_t)(ka + 1) * D_OUT + col];
      acc2[nt] = __builtin_amdgcn_wmma_f32_16x16x4_f32(
          false, a, false, b, (short)0, acc2[nt], false, false);
    }
  }

  // ---- Epilogue 2: + b2, store -------------------------------------------
#pragma unroll
  for (int nt = 0; nt < NT2; ++nt) {
    const int col = nt * 16 + m;
    const float bb = b2[col];
#pragma unroll
    for (int v = 0; v < 8; ++v) {
      const int r = row0 + v + 8 * kh;
      if (r < n_nodes) out[(size_t)r * D_OUT + col] = acc2[nt][v] + bb;
    }
  }
}

// ---------------------------------------------------------------------------
// Pooling + readout
// ---------------------------------------------------------------------------

__global__ void gin_pool(const float* __restrict__ xf,
                         const int* __restrict__ batch,
                         float* __restrict__ pooled,  // [G,16]
                         float* __restrict__ counts,  // [G]
                         int n_nodes) {
  long long tid = (long long)blockIdx.x * blockDim.x + threadIdx.x;
  if (tid >= (long long)n_nodes * 16) return;
  const int i = (int)(tid >> 4);
  const int j = (int)(tid & 15);
  const int g = batch[i];
  atomicAdd(&pooled[g * 16 + j], xf[(size_t)i * 16 + j]);
  if (j == 0) atomicAdd(&counts[g], 1.0f);
}

__global__ void gin_readout(const float* __restrict__ pooled,
                            const float* __restrict__ counts,
                            const float* __restrict__ lin_w,  // [16]
                            const float* __restrict__ lin_b,  // [1]
                            float* __restrict__ out, int n_graphs) {
  int g = blockIdx.x * blockDim.x + threadIdx.x;
  if (g >= n_graphs) return;
  const float inv = 1.0f / fmaxf(counts[g], 1.0f);
  float s = lin_b[0];
#pragma unroll
  for (int j = 0; j < 16; ++j) s += pooled[g * 16 + j] * inv * lin_w[j];
  out[g] = 1.0f / (1.0f + expf(-s));
}

// ---------------------------------------------------------------------------
// Host-side launcher
// ---------------------------------------------------------------------------
extern "C" void kernel_launch(void* const* d_in, const int* in_sizes, int n_in,
                              void* d_out, int out_size, void* d_ws,
                              size_t ws_size, hipStream_t stream) {
  const float* x0 = (const float*)d_in[0];
  const int* ei = (const int*)d_in[1];
  const int* batch = (const int*)d_in[2];
  const int E = in_sizes[1] / 2;
  const int N = in_sizes[2];
  const int G = out_size;
  const int* src = ei;
  const int* dst = ei + E;

  // params (JAX tree flatten = sorted dict keys):
  // per layer: W1,W2,b1,b2,beta,eps,gamma,mean,var ; then lin_b, lin_w
  auto P = [&](int l, int j) { return (const float*)d_in[3 + 9 * l + j]; };
  const float* lin_b = (const float*)d_in[3 + 9 * 4 + 0];
  const float* lin_w = (const float*)d_in[3 + 9 * 4 + 1];

  // ---- workspace partition -----------------------------------------------
  int* iw = (int*)d_ws;
  int* deg = iw;               // N
  int* rowptr = deg + N;       // N+1
  int* partials = rowptr + N + 1;  // 256
  int* cursor = partials + 256;    // N
  int* csr = cursor + N;           // E
  size_t int_total = (size_t)3 * N + 257 + E;
  int_total = (int_total + 3) & ~(size_t)3;  // 16B-align the float region

  float* fw = (float*)d_ws + int_total;
  float* hbuf = fw;                          // N*128
  float* xA = hbuf + (size_t)N * 128;        // N*32
  float* xB = xA + (size_t)N * 32;           // N*32
  float* pooled = xB + (size_t)N * 32;       // G*16
  float* counts = pooled + (size_t)G * 16;   // G

  const int TPB = 256;
  const int tiles = (N + 15) / 16;
  const int nScanBlk = (N + 255) / 256;  // 196 <= 256

  // ---- CSR build (once per call, reused by all 4 layers) ------------------
  gin_zero_i<<<(N + TPB - 1) / TPB, TPB, 0, stream>>>(deg, N);
  gin_degree<<<(E + TPB - 1) / TPB, TPB, 0, stream>>>(dst, deg, E);
  gin_scan_block<<<nScanBlk, 256, 0, stream>>>(deg, rowptr, partials, N);
  gin_scan_partials<<<1, 256, 0, stream>>>(partials, nScanBlk);
  gin_scan_finalize<<<(N + TPB - 1) / TPB, TPB, 0, stream>>>(rowptr, partials,
                                                             cursor, N, E);
  gin_csr_fill<<<(E + TPB - 1) / TPB, TPB, 0, stream>>>(src, dst, cursor, csr,
                                                        E);

  // ---- 4 GIN layers -------------------------------------------------------
  const float* cur = x0;
  float* nxt[4] = {xA, xB, xA, xB};
  const int aggBlk = (N + 3) / 4;  // 4 waves/block, 1 node/wave

  for (int l = 0; l < 4; ++l) {
    if (l == 0)
      gin_aggregate<128><<<aggBlk, 128, 0, stream>>>(cur, rowptr, csr, P(l, 5),
                                                     hbuf, N);
    else
      gin_aggregate<32><<<aggBlk, 128, 0, stream>>>(cur, rowptr, csr, P(l, 5),
                                                    hbuf, N);

    const float *W1 = P(l, 0), *W2 = P(l, 1), *b1 = P(l, 2), *b2 = P(l, 3);
    const float *beta = P(l, 4), *gamma = P(l, 6), *mean = P(l, 7),
                *var = P(l, 8);
    switch (l) {
      case 0:
        gin_mlp<128, 32, 32><<<tiles, 32, 0, stream>>>(
            hbuf, W1, b1, gamma, beta, mean, var, W2, b2, nxt[l], N);
        break;
      case 1:
        gin_mlp<32, 64, 32><<<tiles, 32, 0, stream>>>(
            hbuf, W1, b1, gamma, beta, mean, var, W2, b2, nxt[l], N);
        break;
      case 2:
        gin_mlp<32, 64, 32><<<tiles, 32, 0, stream>>>(
            hbuf, W1, b1, gamma, beta, mean, var, W2, b2, nxt[l], N);
        break;
      default:
        gin_mlp<32, 64, 16><<<tiles, 32, 0, stream>>>(
            hbuf, W1, b1, gamma, beta, mean, var, W2, b2, nxt[l], N);
        break;
    }
    cur = nxt[l];
  }

  // ---- mean pool + readout ------------------------------------------------
  const long long pz4 = (G * 16 + G) / 4;
  gin_zero_f4<<<(int)((pz4 + TPB - 1) / TPB), TPB, 0, stream>>>(
      (float4*)pooled, pz4);
  const long long pn = (long long)N * 16;
  gin_pool<<<(int)((pn + TPB - 1) / TPB), TPB, 0, stream>>>(cur, batch, pooled,
                                                            counts, N);
  gin_readout<<<(G + TPB - 1) / TPB, TPB, 0, stream>>>(pooled, counts, lin_w,
                                                       lin_b, (float*)d_out, G);
}